// NeighborhoodGNN_90486370992782
// MI455X (gfx1250) — compile-verified
//
#include <hip/hip_runtime.h>
#include <hip/hip_bf16.h>
#include <math.h>

typedef __attribute__((ext_vector_type(2))) float v2f;
typedef __attribute__((ext_vector_type(8))) float v8f;

#define HID 64
#define GAT_NEG_SLOPE 0.2f

// ---------------------------------------------------------------------------
// GEMM: C[M x 64] = A[M x K] * W[K x 64] via V_WMMA_F32_16X16X4_F32.
// One wave per 16x16 output tile. A-frag layout per ISA (16x4 f32):
//   lanes 0-15 : M=lane,   VGPR0=K+0, VGPR1=K+1
//   lanes 16-31: M=lane-16,VGPR0=K+2, VGPR1=K+3
// B-frag mirrors A with N in place of M. C/D: VGPR r -> row r (lanes 0-15)
// or row r+8 (lanes 16-31), col = lane%16.
// ---------------------------------------------------------------------------
__global__ void gat_gemm_wmma(const float* __restrict__ A,
                              const float* __restrict__ W,
                              float* __restrict__ C,
                              int M, int K) {
  const int lane = threadIdx.x & 31;
  const int wave = threadIdx.x >> 5;
  const int nTilesN = HID / 16;                       // 4
  const int nTiles  = ((M + 15) / 16) * nTilesN;
  const int tile = blockIdx.x * (blockDim.x >> 5) + wave;
  if (tile >= nTiles) return;
  const int m0 = (tile / nTilesN) * 16;
  const int n0 = (tile % nTilesN) * 16;

  const int half = lane >> 4;    // 0 = lanes 0-15, 1 = lanes 16-31
  const int l16  = lane & 15;

  int arowIdx = m0 + l16;
  if (arowIdx >= M) arowIdx = M - 1;                  // safe clamp (M%16==0 normally)
  const float* arow = A + (size_t)arowIdx * K;

  v8f c = {};
  for (int k = 0; k < K; k += 4) {
    v2f a, b;
    a.x = arow[k + 2 * half + 0];
    a.y = arow[k + 2 * half + 1];
    b.x = W[(size_t)(k + 2 * half + 0) * HID + n0 + l16];
    b.y = W[(size_t)(k + 2 * half + 1) * HID + n0 + l16];
    c = __builtin_amdgcn_wmma_f32_16x16x4_f32(false, a, false, b,
                                              (short)0, c, false, false);
  }

#pragma unroll
  for (int r = 0; r < 8; ++r) {
    int row = m0 + half * 8 + r;
    if (row < M) C[(size_t)row * HID + n0 + l16] = c[r];
  }
}

// ---------------------------------------------------------------------------
// Per-node attention scores: s[i] = h[i].a_src, d[i] = h[i].a_dst  (HID=64)
// One wave per node; wave32 shuffle reduction.
// ---------------------------------------------------------------------------
__global__ void gat_scores(const float* __restrict__ h,
                           const float* __restrict__ a_src,
                           const float* __restrict__ a_dst,
                           float* __restrict__ s, float* __restrict__ d,
                           int N) {
  const int lane = threadIdx.x & 31;
  const int node = blockIdx.x * (blockDim.x >> 5) + (threadIdx.x >> 5);
  if (node >= N) return;
  const float* hr = h + (size_t)node * HID;
  float h0 = hr[lane], h1 = hr[lane + 32];
  float vs = h0 * a_src[lane] + h1 * a_src[lane + 32];
  float vd = h0 * a_dst[lane] + h1 * a_dst[lane + 32];
#pragma unroll
  for (int m = 16; m >= 1; m >>= 1) {
    vs += __shfl_xor(vs, m, 32);
    vd += __shfl_xor(vd, m, 32);
  }
  if (lane == 0) { s[node] = vs; d[node] = vd; }
}

// ---------------------------------------------------------------------------
// Monotone float <-> sortable-uint encoding (for integer atomicMax on floats)
// ---------------------------------------------------------------------------
__device__ __forceinline__ unsigned f2sort(float x) {
  unsigned u = __float_as_uint(x);
  return (u & 0x80000000u) ? ~u : (u | 0x80000000u);
}
__device__ __forceinline__ float sort2f(unsigned u) {
  return __uint_as_float((u & 0x80000000u) ? (u ^ 0x80000000u) : ~u);
}

__global__ void gat_init(float* __restrict__ agg,
                         float* __restrict__ sumb,
                         unsigned* __restrict__ maxb, int N) {
  const int i = blockIdx.x * blockDim.x + threadIdx.x;
  if (i < N * HID) agg[i] = 0.0f;
  if (i < N) { sumb[i] = 0.0f; maxb[i] = 0u; }       // 0u < sortable(any finite)
}

__device__ __forceinline__ void edge_pair(const int* __restrict__ ei,
                                          int E, int idx, int& u, int& v) {
  if (idx < E) { u = ei[idx]; v = ei[E + idx]; }
  else         { u = idx - E; v = idx - E; }          // self-loop
}

__global__ void gat_edge_max(const int* __restrict__ ei,
                             const float* __restrict__ s,
                             const float* __restrict__ d,
                             float* __restrict__ ew,
                             unsigned* __restrict__ maxb, int E, int N) {
  const int idx = blockIdx.x * blockDim.x + threadIdx.x;
  if (idx >= E + N) return;
  int u, v; edge_pair(ei, E, idx, u, v);
  float e = s[u] + d[v];
  e = (e > 0.0f) ? e : GAT_NEG_SLOPE * e;             // leaky relu
  ew[idx] = e;
  atomicMax(maxb + v, f2sort(e));
}

__global__ void gat_edge_expsum(const int* __restrict__ ei,
                                float* __restrict__ ew,
                                const unsigned* __restrict__ maxb,
                                float* __restrict__ sumb, int E, int N) {
  const int idx = blockIdx.x * blockDim.x + threadIdx.x;
  if (idx >= E + N) return;
  int u, v; edge_pair(ei, E, idx, u, v);
  float w = __expf(ew[idx] - sort2f(maxb[v]));
  ew[idx] = w;
  atomicAdd(sumb + v, w);
}

// wave per edge: lane handles features {lane, lane+32}
__global__ void gat_edge_agg(const int* __restrict__ ei,
                             const float* __restrict__ ew,
                             const float* __restrict__ sumb,
                             const float* __restrict__ h,
                             float* __restrict__ agg, int E, int N) {
  const int lane = threadIdx.x & 31;
  const int idx  = blockIdx.x * (blockDim.x >> 5) + (threadIdx.x >> 5);
  if (idx >= E + N) return;
  int u, v; edge_pair(ei, E, idx, u, v);
  const float coef = ew[idx] / sumb[v];
  const float* hr = h + (size_t)u * HID;
  float* ar = agg + (size_t)v * HID;
  atomicAdd(ar + lane,      coef * hr[lane]);
  atomicAdd(ar + lane + 32, coef * hr[lane + 32]);
}

__global__ void gat_finalize(const float* __restrict__ agg,
                             const float* __restrict__ b,
                             float* __restrict__ out, int N) {
  const int i = blockIdx.x * blockDim.x + threadIdx.x;
  if (i >= N * HID) return;
  float v = agg[i] + b[i & (HID - 1)];
  out[i] = v > 0.0f ? v : 0.0f;
}

// ---------------------------------------------------------------------------
extern "C" void kernel_launch(void* const* d_in, const int* in_sizes, int n_in,
                              void* d_out, int out_size, void* d_ws, size_t ws_size,
                              hipStream_t stream) {
  const float* x   = (const float*)d_in[0];
  const int*   ei  = (const int*)d_in[1];
  const float* W1  = (const float*)d_in[2];
  const float* as1 = (const float*)d_in[3];
  const float* ad1 = (const float*)d_in[4];
  const float* b1  = (const float*)d_in[5];
  const float* W2  = (const float*)d_in[6];
  const float* as2 = (const float*)d_in[7];
  const float* ad2 = (const float*)d_in[8];
  const float* b2  = (const float*)d_in[9];

  const int IN = 128;
  const int N  = in_sizes[0] / IN;
  const int E  = in_sizes[1] / 2;
  const int EN = E + N;
  float* out = (float*)d_out;

  // workspace layout
  char* ws = (char*)d_ws;
  float* bufA = (float*)ws; ws += (size_t)N * HID * sizeof(float);  // h (pre-act features)
  float* bufB = (float*)ws; ws += (size_t)N * HID * sizeof(float);  // layer-1 output / agg
  float* sb   = (float*)ws; ws += (size_t)N * sizeof(float);
  float* db   = (float*)ws; ws += (size_t)N * sizeof(float);
  float* sumb = (float*)ws; ws += (size_t)N * sizeof(float);
  unsigned* maxb = (unsigned*)ws; ws += (size_t)N * sizeof(unsigned);
  float* ewb  = (float*)ws;                                          // E+N edge weights

  const int TB = 256;
  const int wavesPerBlock = TB / 32;
  const int nTiles   = ((N + 15) / 16) * (HID / 16);
  const int gGemm    = (nTiles + wavesPerBlock - 1) / wavesPerBlock;
  const int gNodeW   = (N + wavesPerBlock - 1) / wavesPerBlock;      // wave/node
  const int gNF      = (N * HID + TB - 1) / TB;                      // thread/(node,feat)
  const int gEdge    = (EN + TB - 1) / TB;                           // thread/edge
  const int gEdgeW   = (EN + wavesPerBlock - 1) / wavesPerBlock;     // wave/edge

  // ---------------- Layer 1 ----------------
  gat_gemm_wmma <<<gGemm,  TB, 0, stream>>>(x, W1, bufA, N, IN);
  gat_scores    <<<gNodeW, TB, 0, stream>>>(bufA, as1, ad1, sb, db, N);
  gat_init      <<<gNF,    TB, 0, stream>>>(bufB, sumb, maxb, N);
  gat_edge_max  <<<gEdge,  TB, 0, stream>>>(ei, sb, db, ewb, maxb, E, N);
  gat_edge_expsum<<<gEdge, TB, 0, stream>>>(ei, ewb, maxb, sumb, E, N);
  gat_edge_agg  <<<gEdgeW, TB, 0, stream>>>(ei, ewb, sumb, bufA, bufB, E, N);
  gat_finalize  <<<gNF,    TB, 0, stream>>>(bufB, b1, bufB, N);      // bufB = relu(agg+b1)

  // ---------------- Layer 2 ----------------
  gat_gemm_wmma <<<gGemm,  TB, 0, stream>>>(bufB, W2, bufA, N, HID);
  gat_scores    <<<gNodeW, TB, 0, stream>>>(bufA, as2, ad2, sb, db, N);
  gat_init      <<<gNF,    TB, 0, stream>>>(out, sumb, maxb, N);
  gat_edge_max  <<<gEdge,  TB, 0, stream>>>(ei, sb, db, ewb, maxb, E, N);
  gat_edge_expsum<<<gEdge, TB, 0, stream>>>(ei, ewb, maxb, sumb, E, N);
  gat_edge_agg  <<<gEdgeW, TB, 0, stream>>>(ei, ewb, sumb, bufA, out, E, N);
  gat_finalize  <<<gNF,    TB, 0, stream>>>(out, b2, out, N);
}